// FacialGNN_75359496175863
// MI455X (gfx1250) — compile-verified
//
#include <hip/hip_runtime.h>
#include <math.h>

#define NGRAPH 2048
#define NPG    64
#define NNODES (NGRAPH * NPG)     // 131072
#define EDGES  (NNODES * 16)      // 2097152
#define FUSION 128
#define EPSV   1e-5f
#define NEG_SLOPE 0.2f

typedef __attribute__((ext_vector_type(2))) float v2f;
typedef __attribute__((ext_vector_type(8))) float v8f;

__device__ __forceinline__ float leakyf(float a) { return a > 0.f ? a : NEG_SLOPE * a; }

__device__ __forceinline__ float atomicMaxF(float* addr, float val) {
    int old = __float_as_int(*addr);
    while (__int_as_float(old) < val) {
        int assumed = old;
        old = atomicCAS((int*)addr, assumed, __float_as_int(val));
        if (old == assumed) break;
    }
    return __int_as_float(old);
}

// ---------------------------------------------------------------- fill
__global__ void k_fill(float* __restrict__ p, float v, int n) {
    int i = blockIdx.x * blockDim.x + threadIdx.x;
    if (i < n) p[i] = v;
}

// ---------------------------- pad layer-0 inputs so GEMM K is a multiple of 4
__global__ void k_pad_x(const float* __restrict__ x, float* __restrict__ xpad) {
    int n = blockIdx.x * blockDim.x + threadIdx.x;
    if (n >= NNODES) return;
    float v[8];
#pragma unroll
    for (int j = 0; j < 5; ++j) v[j] = x[n * 5 + j];
    v[5] = v[6] = v[7] = 0.f;
#pragma unroll
    for (int j = 0; j < 8; ++j) xpad[n * 8 + j] = v[j];
}

__global__ void k_pad_w0(const float* __restrict__ w, float* __restrict__ wpad) {
    int t = blockIdx.x * blockDim.x + threadIdx.x;   // 8*32
    if (t >= 8 * 32) return;
    int k = t / 32, c = t % 32;
    wpad[t] = (k < 5) ? w[k * 32 + c] : 0.f;
}

// ------------------------------------------- degree + self-loop edge attr
__global__ void k_deg_loop(const int* __restrict__ dst, const float* __restrict__ eattr,
                           float* __restrict__ deg, float* __restrict__ loopat) {
    int e = blockIdx.x * blockDim.x + threadIdx.x;
    if (e >= EDGES) return;
    int d = dst[e];
    atomicAdd(&deg[d], 1.f);
#pragma unroll
    for (int j = 0; j < 6; ++j) atomicAdd(&loopat[d * 6 + j], eattr[e * 6 + j]);
}

__global__ void k_loop_div(float* __restrict__ loopat, const float* __restrict__ deg) {
    int n = blockIdx.x * blockDim.x + threadIdx.x;
    if (n >= NNODES) return;
    float dv = fmaxf(deg[n], 1.f);
#pragma unroll
    for (int j = 0; j < 6; ++j) loopat[n * 6 + j] /= dv;
}

// --------------------------- we_att[6][H] = contract(We, att_edge) (edge-score vector)
__global__ void k_we_att(const float* __restrict__ We, const float* __restrict__ att_edge,
                         float* __restrict__ we_att, int H, int C) {
    int t = threadIdx.x;                  // t = j*H + h
    if (t >= 6 * H) return;
    int j = t / H, h = t % H;
    float s = 0.f;
    for (int c = 0; c < C; ++c) s += We[j * (H * C) + h * C + c] * att_edge[h * C + c];
    we_att[j * H + h] = s;
}

// ---------------------------------------------------------------- WMMA GEMM
// Y[rows, OC] = X[rows, ldx] @ W[IC, OC] (+ bias). One wave = one 16x16 tile.
// IC is a template constant (multiple of 4) -> fully unrolled, guard-free loads,
// EXEC statically all-ones (WMMA requirement). A pair loaded as one b64.
template <int IC>
__global__ void k_gemm_wmma_f32(const float* __restrict__ X, int ldx,
                                const float* __restrict__ W,
                                const float* __restrict__ bias,
                                float* __restrict__ Y, int ldy,
                                int OC, int tilesN) {
    int wave = (blockIdx.x * blockDim.x + threadIdx.x) >> 5;
    int lane = threadIdx.x & 31;
    int tm = wave / tilesN;
    int tn = wave % tilesN;
    int col = tn * 16 + (lane & 15);
    int khalf = (lane >> 4) << 1;         // lanes 0-15: K+0,1 ; lanes 16-31: K+2,3
    const float* xp = X + (tm * 16 + (lane & 15)) * (size_t)ldx + khalf;
    const float* wp = W + (size_t)khalf * OC + col;
    v8f acc = {0.f, 0.f, 0.f, 0.f, 0.f, 0.f, 0.f, 0.f};
#pragma unroll
    for (int k0 = 0; k0 < IC; k0 += 4) {
        v2f a = *(const v2f*)(xp + k0);               // global_load_b64
        v2f b;
        b.x = wp[(size_t)k0 * OC];
        b.y = wp[(size_t)(k0 + 1) * OC];
        acc = __builtin_amdgcn_wmma_f32_16x16x4_f32(false, a, false, b,
                                                    (short)0, acc, false, false);
    }
    int rbase = tm * 16 + ((lane >> 4) << 3);         // C/D: lanes 16-31 hold M+8
    float bc = bias ? bias[col] : 0.f;
#pragma unroll
    for (int v = 0; v < 8; ++v) Y[(rbase + v) * (size_t)ldy + col] = acc[v] + bc;
}

static void launch_gemm(int IC, const float* X, int ldx, const float* W, const float* bias,
                        float* Y, int ldy, int OC, int rows, hipStream_t stream) {
    int tilesN = OC / 16;
    int blocks = (rows / 16) * tilesN / 8;            // 8 waves (tiles) per block
    switch (IC) {
        case 8:   k_gemm_wmma_f32<8><<<blocks, 256, 0, stream>>>(X, ldx, W, bias, Y, ldy, OC, tilesN); break;
        case 32:  k_gemm_wmma_f32<32><<<blocks, 256, 0, stream>>>(X, ldx, W, bias, Y, ldy, OC, tilesN); break;
        case 64:  k_gemm_wmma_f32<64><<<blocks, 256, 0, stream>>>(X, ldx, W, bias, Y, ldy, OC, tilesN); break;
        default:  k_gemm_wmma_f32<128><<<blocks, 256, 0, stream>>>(X, ldx, W, bias, Y, ldy, OC, tilesN); break;
    }
}

// ----------------- per-node attention scalars; amax seeded with self-loop alpha
__global__ void k_node_att(const float* __restrict__ xh, const float* __restrict__ att_src,
                           const float* __restrict__ att_dst, const float* __restrict__ loopat,
                           const float* __restrict__ we_att,
                           float* __restrict__ asrc, float* __restrict__ adst,
                           float* __restrict__ aself, float* __restrict__ amax,
                           float* __restrict__ denom, int H, int C) {
    int t = blockIdx.x * blockDim.x + threadIdx.x;
    if (t >= NNODES * H) return;
    int n = t / H, h = t % H, HC = H * C;
    float ss = 0.f, sd = 0.f;
    for (int c = 0; c < C; ++c) {
        float xv = xh[n * HC + h * C + c];
        ss += xv * att_src[h * C + c];
        sd += xv * att_dst[h * C + c];
    }
    asrc[t] = ss; adst[t] = sd;
    float ae = 0.f;
    for (int j = 0; j < 6; ++j) ae += loopat[n * 6 + j] * we_att[j * H + h];
    float al = leakyf(ss + sd + ae);
    aself[t] = al;
    amax[t]  = al;     // self-loop participates in the per-destination max
    denom[t] = 0.f;
}

// --------------------------------------- pass 1 over edges: running max of alpha
__global__ void k_edge_max(const int* __restrict__ src, const int* __restrict__ dst,
                           const float* __restrict__ eattr, const float* __restrict__ we_att,
                           const float* __restrict__ asrc, const float* __restrict__ adst,
                           float* __restrict__ amax, int H) {
    int e = blockIdx.x * blockDim.x + threadIdx.x;
    if (e >= EDGES) return;
    int s = src[e], d = dst[e];
    float ea[6];
#pragma unroll
    for (int j = 0; j < 6; ++j) ea[j] = eattr[e * 6 + j];
    for (int h = 0; h < H; ++h) {
        float ae = 0.f;
#pragma unroll
        for (int j = 0; j < 6; ++j) ae += ea[j] * we_att[j * H + h];
        float al = leakyf(asrc[s * H + h] + adst[d * H + h] + ae);
        atomicMaxF(&amax[d * H + h], al);
    }
}

// ----------- pass 2: exp, denom accumulation, unnormalized weighted feature scatter
__global__ void k_edge_aggr(const int* __restrict__ src, const int* __restrict__ dst,
                            const float* __restrict__ eattr, const float* __restrict__ we_att,
                            const float* __restrict__ asrc, const float* __restrict__ adst,
                            const float* __restrict__ amax, float* __restrict__ denom,
                            const float* __restrict__ xh, float* __restrict__ aggr,
                            int H, int C) {
    int e = blockIdx.x * blockDim.x + threadIdx.x;
    if (e >= EDGES) return;
    int s = src[e], d = dst[e], HC = H * C;
    float ea[6];
#pragma unroll
    for (int j = 0; j < 6; ++j) ea[j] = eattr[e * 6 + j];
    for (int h = 0; h < H; ++h) {
        float ae = 0.f;
#pragma unroll
        for (int j = 0; j < 6; ++j) ae += ea[j] * we_att[j * H + h];
        float al = leakyf(asrc[s * H + h] + adst[d * H + h] + ae);
        float ex = __expf(al - amax[d * H + h]);
        atomicAdd(&denom[d * H + h], ex);
        for (int c = 0; c < C; ++c)
            atomicAdd(&aggr[d * HC + h * C + c], ex * xh[s * HC + h * C + c]);
    }
}

// ------ per-graph block: softmax finalize (+self loop) + bias + graph-LN + ReLU + pool
__global__ void k_finalize_ln_pool(const float* __restrict__ xh, float* __restrict__ aggr,
                                   const float* __restrict__ aself, const float* __restrict__ amax,
                                   const float* __restrict__ denom, const float* __restrict__ bias,
                                   const float* __restrict__ lnw, const float* __restrict__ lnb,
                                   float* __restrict__ gpool, int H, int C) {
    __shared__ float tile[NPG * 128];
    __shared__ float red[256];
    int g = blockIdx.x, tid = threadIdx.x;
    int HC = H * C, total = NPG * HC;
    float lsum = 0.f, lsq = 0.f;
    for (int t = tid; t < total; t += blockDim.x) {
        int nl = t / HC, f = t % HC, h = f / C;
        int n = g * NPG + nl;
        float es = __expf(aself[n * H + h] - amax[n * H + h]);
        float dt = denom[n * H + h] + es;
        float val = (aggr[n * HC + f] + es * xh[n * HC + f]) / dt + bias[f];
        tile[t] = val;
        lsum += val; lsq += val * val;
    }
    red[tid] = lsum; __syncthreads();
    for (int s = 128; s > 0; s >>= 1) { if (tid < s) red[tid] += red[tid + s]; __syncthreads(); }
    float sum = red[0]; __syncthreads();
    red[tid] = lsq; __syncthreads();
    for (int s = 128; s > 0; s >>= 1) { if (tid < s) red[tid] += red[tid + s]; __syncthreads(); }
    float sq = red[0]; __syncthreads();
    float cnt = (float)total;
    float mean = sum / cnt;
    float var = fmaxf(sq / cnt - mean * mean, 0.f);
    float inv = rsqrtf(var + EPSV);
    for (int t = tid; t < total; t += blockDim.x) {
        int nl = t / HC, f = t % HC;
        float y = lnw[f] * (tile[t] - mean) * inv + lnb[f];
        y = fmaxf(y, 0.f);
        tile[t] = y;
        aggr[(g * NPG + nl) * HC + f] = y;   // becomes next layer's node features
    }
    __syncthreads();
    for (int f = tid; f < HC; f += blockDim.x) {
        float s = 0.f;
        for (int nl = 0; nl < NPG; ++nl) s += tile[nl * HC + f];
        gpool[g * HC + f] = s * (1.f / NPG);
    }
}

// ------------------------------- gated fusion: out += sigmoid(gs·am_w + am_b) * gs
__global__ void k_gate_add(const float* __restrict__ gsi, const float* __restrict__ amw,
                           const float* __restrict__ amb, float* __restrict__ out) {
    __shared__ float red[128];
    int g = blockIdx.x, f = threadIdx.x;
    float v = gsi[g * FUSION + f];
    red[f] = v * amw[f]; __syncthreads();
    for (int s = 64; s > 0; s >>= 1) { if (f < s) red[f] += red[f + s]; __syncthreads(); }
    float a = 1.f / (1.f + __expf(-(red[0] + amb[0])));
    out[g * FUSION + f] += a * v;
}

// ================================================================ launch
extern "C" void kernel_launch(void* const* d_in, const int* in_sizes, int n_in,
                              void* d_out, int out_size, void* d_ws, size_t ws_size,
                              hipStream_t stream) {
    (void)in_sizes; (void)n_in; (void)out_size; (void)ws_size;
    const float* x     = (const float*)d_in[0];
    const int*   ei    = (const int*)d_in[1];
    const float* eattr = (const float*)d_in[3];
    const int* src = ei;
    const int* dst = ei + EDGES;
    auto P = [&](int layer, int idx) { return (const float*)d_in[4 + layer * 12 + idx]; };

    const int ICs[3] = {5, 32, 64};
    const int HCs[3] = {32, 64, 128};
    const int Hs[3]  = {4, 4, 1};

    // workspace layout (floats)
    float* ws     = (float*)d_ws;
    float* deg    = ws;                         // N
    float* loopat = deg    + NNODES;            // 6N
    float* we_att = loopat + NNODES * 6;        // 32
    float* wpad   = we_att + 32;                // 8*32
    float* asrc   = wpad   + 8 * 32;            // 4N
    float* adst   = asrc   + NNODES * 4;        // 4N
    float* aself  = adst   + NNODES * 4;        // 4N
    float* amax   = aself  + NNODES * 4;        // 4N
    float* denom  = amax   + NNODES * 4;        // 4N
    float* gpool  = denom  + NNODES * 4;        // G*128
    float* gs     = gpool  + NGRAPH * 128;      // 3*G*128
    float* xpad   = gs     + 3 * NGRAPH * 128;  // N*8 (padded layer-0 input)
    float* xbuf   = xpad   + NNODES * 8;        // N*128 (node feats; aliases aggr)
    float* xh     = xbuf   + NNODES * 128;      // N*128
    float* out    = (float*)d_out;

    const int T = 256;
    // layer-0 K-padding + self-loop attr precompute (layer-independent)
    k_pad_x<<<NNODES / T, T, 0, stream>>>(x, xpad);
    k_pad_w0<<<1, 256, 0, stream>>>(P(0, 0), wpad);
    k_fill<<<(NNODES + T - 1) / T, T, 0, stream>>>(deg, 0.f, NNODES);
    k_fill<<<(NNODES * 6 + T - 1) / T, T, 0, stream>>>(loopat, 0.f, NNODES * 6);
    k_deg_loop<<<EDGES / T, T, 0, stream>>>(dst, eattr, deg, loopat);
    k_loop_div<<<NNODES / T, T, 0, stream>>>(loopat, deg);

    for (int i = 0; i < 3; ++i) {
        int HC = HCs[i], H = Hs[i], C = HC / H;
        const float* Xin = (i == 0) ? xpad : xbuf;
        const float* Wl  = (i == 0) ? wpad : P(i, 0);
        int IC  = (i == 0) ? 8 : ICs[i];
        int ldx = (i == 0) ? 8 : HCs[i - 1];

        k_we_att<<<1, 32, 0, stream>>>(P(i, 3), P(i, 4), we_att, H, C);

        launch_gemm(IC, Xin, ldx, Wl, nullptr, xh, HC, HC, NNODES, stream);

        k_fill<<<(NNODES * HC + T - 1) / T, T, 0, stream>>>(xbuf, 0.f, NNODES * HC); // aggr
        k_node_att<<<(NNODES * H + T - 1) / T, T, 0, stream>>>(
            xh, P(i, 1), P(i, 2), loopat, we_att, asrc, adst, aself, amax, denom, H, C);
        k_edge_max<<<EDGES / T, T, 0, stream>>>(src, dst, eattr, we_att, asrc, adst, amax, H);
        k_edge_aggr<<<EDGES / T, T, 0, stream>>>(src, dst, eattr, we_att, asrc, adst,
                                                 amax, denom, xh, xbuf, H, C);
        k_finalize_ln_pool<<<NGRAPH, 256, 0, stream>>>(
            xh, xbuf, aself, amax, denom, P(i, 5), P(i, 6), P(i, 7), gpool, H, C);

        // gs[i] = gpool @ proj_w + proj_b  (WMMA, bias fused)
        launch_gemm(HC, gpool, HC, P(i, 8), P(i, 9),
                    gs + i * NGRAPH * FUSION, FUSION, FUSION, NGRAPH, stream);
    }

    k_fill<<<(NGRAPH * FUSION + T - 1) / T, T, 0, stream>>>(out, 0.f, NGRAPH * FUSION);
    for (int i = 0; i < 3; ++i)
        k_gate_add<<<NGRAPH, 128, 0, stream>>>(gs + i * NGRAPH * FUSION, P(i, 10), P(i, 11), out);
}